// DensityLoss_47236050321867
// MI455X (gfx1250) — compile-verified
//
#include <hip/hip_runtime.h>

// ---------------------------------------------------------------------------
// DensityLoss on MI455X (gfx1250, wave32).
//
// Algebraic reduction of the reference:
//   s       = sum_j centers[j]                (column sum, [D])
//   sq[i]   = centers[i] . centers[i]
//   S       = sum_i sq[i]
//   cd[i]   = (C*sq[i] + S - 2*centers[i].s) / (C-1)     (diag of dist == 0)
//   out     = mean(cd) / var_unbiased(cd) / N
//
// O(C*D) = 4MB of data -> memory-bound (~0.4us at 23.3 TB/s).
// sq[] and p[] = centers . s are computed with chained V_WMMA_F32_16X16X4_F32:
// per 16-row tile, A x A^T accumulates the 16x16 gram block (diag = sq) and
// A x [s s ... s] gives p in every column. The A-fragment doubles as the
// B-fragment for A^T (lane m == lane n), so one b64 load feeds both chains.
// ---------------------------------------------------------------------------

typedef __attribute__((ext_vector_type(2))) float v2f;
typedef __attribute__((ext_vector_type(8))) float v8f;

#define C_DIM 4096
#define D_DIM 256
#define P1_ROWS 64
#define P1_BLOCKS (C_DIM / P1_ROWS) // 64

// ---- Phase 1a: partial column sums: partial[b][t] = sum over 64 rows of col t
__global__ void __launch_bounds__(256) dl_colsum_partial(
    const float* __restrict__ centers, float* __restrict__ partial) {
  const int t = threadIdx.x;               // column 0..255
  const int b = blockIdx.x;                // row-chunk 0..63
  const float* rp = centers + (size_t)b * P1_ROWS * D_DIM + t;
  float acc = 0.0f;
#pragma unroll 8
  for (int r = 0; r < P1_ROWS; ++r) acc += rp[r * D_DIM];
  partial[b * D_DIM + t] = acc;
}

// ---- Phase 1b: s[t] = sum_b partial[b][t]   (fixed order -> deterministic)
__global__ void __launch_bounds__(256) dl_colsum_final(
    const float* __restrict__ partial, float* __restrict__ s) {
  const int t = threadIdx.x;
  float acc = 0.0f;
#pragma unroll 8
  for (int b = 0; b < P1_BLOCKS; ++b) acc += partial[b * D_DIM + t];
  s[t] = acc;
}

// ---- Phase 2: per 16-row tile, WMMA f32 chains for sq[] and p[].
// Block = 256 threads = 8 waves; each wave owns one 16-row tile.
// Grid = 32 blocks -> 256 tiles * 16 rows = 4096 rows.
__global__ void __launch_bounds__(256) dl_rowdot_wmma(
    const float* __restrict__ centers, const float* __restrict__ s,
    float* __restrict__ sq, float* __restrict__ p) {
  const int lane = threadIdx.x & 31;
  const int wave = threadIdx.x >> 5;
  const int rowbase = (blockIdx.x * 8 + wave) * 16;

  // A-fragment addressing (16x4 f32): lanes 0-15 -> K={0,1}, lanes 16-31 -> K={2,3}
  const int m = lane & 15;
  const int kbase = (lane >> 4) << 1; // 0 or 2
  const float* aptr = centers + (size_t)(rowbase + m) * D_DIM + kbase;

  v8f cg = {}; // accumulates 16x16 gram block of this tile (diag = sq)
  v8f cp = {}; // accumulates A x [s s .. s]      (every column = p)

#pragma unroll 4
  for (int k = 0; k < D_DIM; k += 4) {
    v2f a = *(const v2f*)(aptr + k);          // A[m][k+kbase .. +1], b64 load
    v2f bs = *(const v2f*)(s + k + kbase);    // s[k+kbase .. +1] (broadcast)
    // gram chain: B = A^T -> B-fragment == A-fragment at lane n == m
    cg = __builtin_amdgcn_wmma_f32_16x16x4_f32(
        /*neg_a=*/false, a, /*neg_b=*/false, a,
        /*c_mod=*/(short)0, cg, /*reuse_a=*/false, /*reuse_b=*/false);
    // p chain: every column of B is s
    cp = __builtin_amdgcn_wmma_f32_16x16x4_f32(
        /*neg_a=*/false, a, /*neg_b=*/false, bs,
        /*c_mod=*/(short)0, cp, /*reuse_a=*/false, /*reuse_b=*/false);
  }

  // C/D layout: VGPR v, lanes 0-15: (M=v, N=lane); lanes 16-31: (M=v+8, N=lane-16).
  // Diagonal element i lives at: i<8 -> lane=i, vgpr=i ; i>=8 -> lane=i+16, vgpr=i-8.
  const bool lo = (lane < 8);
  const bool hi = (lane >= 24);
  const int v_sel = lo ? lane : (lane - 24); // 0..7 when lo||hi
  float sqv = 0.0f, pv = 0.0f;
#pragma unroll
  for (int v = 0; v < 8; ++v) {
    if ((lo || hi) && v_sel == v) { sqv = cg[v]; pv = cp[v]; }
  }
  if (lo || hi) {
    const int i = lo ? lane : (lane - 16); // diag index 0..15
    sq[rowbase + i] = sqv;
    p[rowbase + i] = pv;
  }
}

// ---- Phase 3: single block, LDS tree reductions; two-pass variance.
__global__ void __launch_bounds__(256) dl_finalize(
    const float* __restrict__ sq, const float* __restrict__ p,
    float* __restrict__ out, float inv_n) {
  __shared__ float red[256];
  __shared__ float bcast;
  const int t = threadIdx.x;
  const float scale = 1.0f / (float)(C_DIM - 1);

  // S = sum sq
  float acc = 0.0f;
#pragma unroll
  for (int i = 0; i < C_DIM / 256; ++i) acc += sq[t + i * 256];
  red[t] = acc;
  __syncthreads();
  for (int off = 128; off > 0; off >>= 1) {
    if (t < off) red[t] += red[t + off];
    __syncthreads();
  }
  if (t == 0) bcast = red[0];
  __syncthreads();
  const float S = bcast;
  __syncthreads();

  // sum of cd
  float acc2 = 0.0f;
#pragma unroll
  for (int i = 0; i < C_DIM / 256; ++i) {
    const int r = t + i * 256;
    acc2 += ((float)C_DIM * sq[r] + S - 2.0f * p[r]) * scale;
  }
  red[t] = acc2;
  __syncthreads();
  for (int off = 128; off > 0; off >>= 1) {
    if (t < off) red[t] += red[t + off];
    __syncthreads();
  }
  if (t == 0) bcast = red[0];
  __syncthreads();
  const float sum_cd = bcast;
  const float mean_cd = sum_cd / (float)C_DIM;
  __syncthreads();

  // sum of (cd - mean)^2  (two-pass: avoids ~9-bit cancellation of E[x^2]-E[x]^2)
  float acc3 = 0.0f;
#pragma unroll
  for (int i = 0; i < C_DIM / 256; ++i) {
    const int r = t + i * 256;
    const float cd = ((float)C_DIM * sq[r] + S - 2.0f * p[r]) * scale;
    const float d = cd - mean_cd;
    acc3 += d * d;
  }
  red[t] = acc3;
  __syncthreads();
  for (int off = 128; off > 0; off >>= 1) {
    if (t < off) red[t] += red[t + off];
    __syncthreads();
  }
  if (t == 0) {
    const float var = red[0] * scale; // unbiased: / (C-1)
    out[0] = (mean_cd / var) * inv_n;
  }
}

// ---------------------------------------------------------------------------
extern "C" void kernel_launch(void* const* d_in, const int* in_sizes, int n_in,
                              void* d_out, int out_size, void* d_ws, size_t ws_size,
                              hipStream_t stream) {
  const float* centers = (const float*)d_in[0];
  // d_in[1] (features) unused by the forward math; d_in[2] (labels) only via N.
  const float inv_n = 1.0f / (float)in_sizes[2];

  float* ws = (float*)d_ws;
  float* partial = ws;                        // 64*256
  float* s_vec   = partial + P1_BLOCKS * D_DIM; // 256
  float* sq      = s_vec + D_DIM;             // 4096
  float* p       = sq + C_DIM;                // 4096

  dl_colsum_partial<<<P1_BLOCKS, 256, 0, stream>>>(centers, partial);
  dl_colsum_final<<<1, 256, 0, stream>>>(partial, s_vec);
  dl_rowdot_wmma<<<C_DIM / 128, 256, 0, stream>>>(centers, s_vec, sq, p);
  dl_finalize<<<1, 256, 0, stream>>>(sq, p, (float*)d_out, inv_n);
}